// AmpPerLoss_82076825026755
// MI455X (gfx1250) — compile-verified
//
#include <hip/hip_runtime.h>
#include <math.h>

typedef __attribute__((ext_vector_type(2))) float v2f;
typedef __attribute__((ext_vector_type(8))) float v8f;

#define B_DIM   128
#define L_DIM   100000
#define BLK_X   32
#define NTHREADS 256
#define NPART   (BLK_X * B_DIM)   // 4096 per-block partials, multiple of 64

static_assert(NPART % 64 == 0, "WMMA reduction needs multiple of 64");

__device__ __forceinline__ float sigmoidf_fast(float x) {
    return 1.0f / (1.0f + __expf(-x));
}

// Monotone fp32 -> uint32 key: unsigned compare == float compare
__device__ __forceinline__ unsigned f2key(float f) {
    unsigned u = __float_as_uint(f);
    return (u & 0x80000000u) ? ~u : (u | 0x80000000u);
}
__device__ __forceinline__ float key2f(unsigned k) {
    unsigned u = (k & 0x80000000u) ? (k & 0x7FFFFFFFu) : ~k;
    return __uint_as_float(u);
}

// ---------------- init: reset per-batch min/max state each call ----------------
__global__ void init_kernel(int* lo_t, int* hi_t, int* lo_p, int* hi_p,
                            unsigned* smin_t, unsigned* smax_t,
                            unsigned* smin_p, unsigned* smax_p) {
    int b = blockIdx.x * blockDim.x + threadIdx.x;
    if (b < B_DIM) {
        lo_t[b] = L_DIM;  hi_t[b] = -1;
        lo_p[b] = L_DIM;  hi_p[b] = -1;
        smin_t[b] = 0xFFFFFFFFu;  smax_t[b] = 0u;
        smin_p[b] = 0xFFFFFFFFu;  smax_p[b] = 0u;
    }
}

// ---------------- pass 1: BCE + smoothness partials, mask index bounds ----------------
__global__ void pass1_kernel(const float* __restrict__ p, const float* __restrict__ t,
                             float* __restrict__ bce_part, float* __restrict__ smooth_part,
                             int* __restrict__ lo_t, int* __restrict__ hi_t,
                             int* __restrict__ lo_p, int* __restrict__ hi_p) {
    const int b   = blockIdx.y;
    const int tid = threadIdx.x;
    const float* pr = p + (size_t)b * L_DIM;
    const float* tr = t + (size_t)b * L_DIM;
    const int nvec    = L_DIM >> 2;
    const int gtid    = blockIdx.x * blockDim.x + tid;
    const int gstride = gridDim.x * blockDim.x;

    float bce = 0.0f, sm = 0.0f;
    int mnT = L_DIM, mxT = -1, mnP = L_DIM, mxP = -1;

    for (int j = gtid; j < nvec; j += gstride) {
        const float4 pv = reinterpret_cast<const float4*>(pr)[j];
        const float4 tv = reinterpret_cast<const float4*>(tr)[j];
        const int i0 = j << 2;
        float pe[5];
        pe[0] = pv.x; pe[1] = pv.y; pe[2] = pv.z; pe[3] = pv.w;
        pe[4] = (i0 + 4 < L_DIM) ? pr[i0 + 4] : 0.0f;   // neighbor for last pair (L2 hit)
        const float te[4] = {tv.x, tv.y, tv.z, tv.w};
        float sg[5];
#pragma unroll
        for (int k = 0; k < 5; ++k) sg[k] = sigmoidf_fast(pe[k]);
#pragma unroll
        for (int k = 0; k < 4; ++k) {
            const float pp = pe[k];
            const int   i  = i0 + k;
            bce += fmaxf(pp, 0.0f) - pp * te[k] + log1pf(__expf(-fabsf(pp)));
            if (te[k] > 0.5f) { mnT = min(mnT, i); mxT = max(mxT, i); }
            if (pp    > 0.0f) { mnP = min(mnP, i); mxP = max(mxP, i); }
            if (i < L_DIM - 1) sm += fabsf(sg[k + 1] - sg[k]);
        }
    }

    __shared__ float rf[NTHREADS];
    __shared__ int   ri[NTHREADS];

    rf[tid] = bce; __syncthreads();
    for (int s2 = NTHREADS / 2; s2 > 0; s2 >>= 1) { if (tid < s2) rf[tid] += rf[tid + s2]; __syncthreads(); }
    if (tid == 0) bce_part[b * gridDim.x + blockIdx.x] = rf[0];
    __syncthreads();

    rf[tid] = sm; __syncthreads();
    for (int s2 = NTHREADS / 2; s2 > 0; s2 >>= 1) { if (tid < s2) rf[tid] += rf[tid + s2]; __syncthreads(); }
    if (tid == 0) smooth_part[b * gridDim.x + blockIdx.x] = rf[0];
    __syncthreads();

    ri[tid] = mnT; __syncthreads();
    for (int s2 = NTHREADS / 2; s2 > 0; s2 >>= 1) { if (tid < s2) ri[tid] = min(ri[tid], ri[tid + s2]); __syncthreads(); }
    if (tid == 0) atomicMin(&lo_t[b], ri[0]);
    __syncthreads();

    ri[tid] = mxT; __syncthreads();
    for (int s2 = NTHREADS / 2; s2 > 0; s2 >>= 1) { if (tid < s2) ri[tid] = max(ri[tid], ri[tid + s2]); __syncthreads(); }
    if (tid == 0) atomicMax(&hi_t[b], ri[0]);
    __syncthreads();

    ri[tid] = mnP; __syncthreads();
    for (int s2 = NTHREADS / 2; s2 > 0; s2 >>= 1) { if (tid < s2) ri[tid] = min(ri[tid], ri[tid + s2]); __syncthreads(); }
    if (tid == 0) atomicMin(&lo_p[b], ri[0]);
    __syncthreads();

    ri[tid] = mxP; __syncthreads();
    for (int s2 = NTHREADS / 2; s2 > 0; s2 >>= 1) { if (tid < s2) ri[tid] = max(ri[tid], ri[tid + s2]); __syncthreads(); }
    if (tid == 0) atomicMax(&hi_p[b], ri[0]);
}

// ---------------- pass 2: windowed min/max of signals per batch ----------------
__global__ void pass2_kernel(const float* __restrict__ s,
                             const int* __restrict__ lo_t, const int* __restrict__ hi_t,
                             const int* __restrict__ lo_p, const int* __restrict__ hi_p,
                             unsigned* __restrict__ smin_t, unsigned* __restrict__ smax_t,
                             unsigned* __restrict__ smin_p, unsigned* __restrict__ smax_p) {
    const int b   = blockIdx.y;
    const int tid = threadIdx.x;
    const int lt = lo_t[b], ht = hi_t[b], lp = lo_p[b], hp = hi_p[b];
    const float* sr = s + (size_t)b * L_DIM;
    const int nvec    = L_DIM >> 2;
    const int gtid    = blockIdx.x * blockDim.x + tid;
    const int gstride = gridDim.x * blockDim.x;

    unsigned mnT = 0xFFFFFFFFu, mxT = 0u, mnP = 0xFFFFFFFFu, mxP = 0u;

    for (int j = gtid; j < nvec; j += gstride) {
        const float4 sv = reinterpret_cast<const float4*>(sr)[j];
        const int i0 = j << 2;
        const float se[4] = {sv.x, sv.y, sv.z, sv.w};
#pragma unroll
        for (int k = 0; k < 4; ++k) {
            const int i = i0 + k;
            const unsigned key = f2key(se[k]);
            if (i >= lt && i <= ht) { mnT = min(mnT, key); mxT = max(mxT, key); }
            if (i >= lp && i <= hp) { mnP = min(mnP, key); mxP = max(mxP, key); }
        }
    }

    __shared__ unsigned ru[NTHREADS];

    ru[tid] = mnT; __syncthreads();
    for (int s2 = NTHREADS / 2; s2 > 0; s2 >>= 1) { if (tid < s2) ru[tid] = min(ru[tid], ru[tid + s2]); __syncthreads(); }
    if (tid == 0) atomicMin(&smin_t[b], ru[0]);
    __syncthreads();

    ru[tid] = mxT; __syncthreads();
    for (int s2 = NTHREADS / 2; s2 > 0; s2 >>= 1) { if (tid < s2) ru[tid] = max(ru[tid], ru[tid + s2]); __syncthreads(); }
    if (tid == 0) atomicMax(&smax_t[b], ru[0]);
    __syncthreads();

    ru[tid] = mnP; __syncthreads();
    for (int s2 = NTHREADS / 2; s2 > 0; s2 >>= 1) { if (tid < s2) ru[tid] = min(ru[tid], ru[tid + s2]); __syncthreads(); }
    if (tid == 0) atomicMin(&smin_p[b], ru[0]);
    __syncthreads();

    ru[tid] = mxP; __syncthreads();
    for (int s2 = NTHREADS / 2; s2 > 0; s2 >>= 1) { if (tid < s2) ru[tid] = max(ru[tid], ru[tid + s2]); __syncthreads(); }
    if (tid == 0) atomicMax(&smax_p[b], ru[0]);
}

// ---------------- final: WMMA-based exact f32 sum of partials + combine ----------------
// One wave32. A = all-ones 16x4 f32 => D = 1*B + C sums 64 new partials per
// v_wmma_f32_16x16x4_f32. Every row of D equals the column sums, so summing one
// D VGPR across all 32 lanes gives exactly 2x the grand total (halved exactly).
__global__ void final_kernel(const float* __restrict__ bce_part,
                             const float* __restrict__ smooth_part,
                             const int* __restrict__ hi_t, const int* __restrict__ hi_p,
                             const unsigned* __restrict__ smin_t, const unsigned* __restrict__ smax_t,
                             const unsigned* __restrict__ smin_p, const unsigned* __restrict__ smax_p,
                             float* __restrict__ out) {
    const int lane = threadIdx.x;   // 32 threads = exactly one wave, EXEC all 1s

    v2f a; a[0] = 1.0f; a[1] = 1.0f;      // all-ones A (16x4)
    v8f cb = {}; v8f cs = {};

    for (int base = 0; base < NPART; base += 64) {
        v2f bb; bb[0] = bce_part[base + lane];    bb[1] = bce_part[base + 32 + lane];
        cb = __builtin_amdgcn_wmma_f32_16x16x4_f32(false, a, false, bb, (short)0, cb, false, false);
        v2f bs; bs[0] = smooth_part[base + lane]; bs[1] = smooth_part[base + 32 + lane];
        cs = __builtin_amdgcn_wmma_f32_16x16x4_f32(false, a, false, bs, (short)0, cs, false, false);
    }

    float x = cb[0];
    for (int off = 16; off > 0; off >>= 1) x += __shfl_xor(x, off, 32);
    const float bce_total = x * 0.5f;

    float y = cs[0];
    for (int off = 16; off > 0; off >>= 1) y += __shfl_xor(y, off, 32);
    const float smooth_total = y * 0.5f;

    // Amp loss: 128 batches across 32 lanes (4 each)
    float aacc = 0.0f;
    for (int b = lane; b < B_DIM; b += 32) {
        if (hi_t[b] >= 0 && hi_p[b] >= 0) {
            const float ta = key2f(smax_t[b]) - key2f(smin_t[b]);
            const float pa = key2f(smax_p[b]) - key2f(smin_p[b]);
            const float d  = fabsf(ta - pa);
            aacc += (ta > 1e-6f) ? d / (ta + 1e-6f) : d;
        }
    }
    for (int off = 16; off > 0; off >>= 1) aacc += __shfl_xor(aacc, off, 32);

    if (lane == 0) {
        const float bce    = bce_total    / ((float)B_DIM * (float)L_DIM);
        const float smooth = smooth_total / ((float)B_DIM * (float)(L_DIM - 1));
        const float amp    = aacc / (float)B_DIM;
        out[0] = 1.0f * bce + 0.5f * amp + 0.3f * smooth;
    }
}

extern "C" void kernel_launch(void* const* d_in, const int* in_sizes, int n_in,
                              void* d_out, int out_size, void* d_ws, size_t ws_size,
                              hipStream_t stream) {
    const float* signals     = (const float*)d_in[0];   // (B,1,L) contiguous
    const float* predictions = (const float*)d_in[1];   // (B,L,1) contiguous
    const float* targets     = (const float*)d_in[2];   // (B,L,1) contiguous
    float* out = (float*)d_out;

    // workspace carve-up (36 KB total)
    int* lo_t = (int*)d_ws;
    int* hi_t = lo_t + B_DIM;
    int* lo_p = hi_t + B_DIM;
    int* hi_p = lo_p + B_DIM;
    unsigned* smin_t = (unsigned*)(hi_p + B_DIM);
    unsigned* smax_t = smin_t + B_DIM;
    unsigned* smin_p = smax_t + B_DIM;
    unsigned* smax_p = smin_p + B_DIM;
    float* bce_part    = (float*)(smax_p + B_DIM);   // NPART
    float* smooth_part = bce_part + NPART;           // NPART

    init_kernel<<<1, 128, 0, stream>>>(lo_t, hi_t, lo_p, hi_p,
                                       smin_t, smax_t, smin_p, smax_p);

    dim3 grid(BLK_X, B_DIM);
    pass1_kernel<<<grid, NTHREADS, 0, stream>>>(predictions, targets,
                                                bce_part, smooth_part,
                                                lo_t, hi_t, lo_p, hi_p);
    pass2_kernel<<<grid, NTHREADS, 0, stream>>>(signals,
                                                lo_t, hi_t, lo_p, hi_p,
                                                smin_t, smax_t, smin_p, smax_p);
    final_kernel<<<1, 32, 0, stream>>>(bce_part, smooth_part,
                                       hi_t, hi_p,
                                       smin_t, smax_t, smin_p, smax_p, out);
}